// CPNet_62680752717907
// MI455X (gfx1250) — compile-verified
//
#include <hip/hip_runtime.h>
#include <stdint.h>

#define BS 256
#define NODE_IPT 4
#define NODE_TILE (BS * NODE_IPT)     // 1024 node ids per block
#define EDGE_IPT 8
#define EDGE_TILE (BS * EDGE_IPT)     // 2048 edges per block
#define XG_WAVES 8                    // 8 wave32s per block, one output row each

// ---------------------------------------------------------------------------
// helpers
// ---------------------------------------------------------------------------
__device__ __forceinline__ int block_exscan256(int v, int* sh, int& total) {
  int tid = threadIdx.x;
  sh[tid] = v;
  __syncthreads();
#pragma unroll
  for (int off = 1; off < BS; off <<= 1) {
    int t = (tid >= off) ? sh[tid - off] : 0;
    __syncthreads();
    sh[tid] += t;
    __syncthreads();
  }
  int incl = sh[tid];
  total = sh[BS - 1];
  return incl - v;
}

__device__ __forceinline__ uint32_t lds_byte_off(const void* p) {
  // LDS aperture places the byte offset in the low 32 bits of the flat address
  return (uint32_t)(uintptr_t)p;
}

// load 8 consecutive edge endpoints; vectorized (2x b128) when aligned
__device__ __forceinline__ void load8(const int* __restrict__ p, bool aligned16,
                                      int v[EDGE_IPT]) {
  if (aligned16) {
    int4 a = *(const int4*)p;
    int4 b = *(const int4*)(p + 4);
    v[0] = a.x; v[1] = a.y; v[2] = a.z; v[3] = a.w;
    v[4] = b.x; v[5] = b.y; v[6] = b.z; v[7] = b.w;
  } else {
#pragma unroll
    for (int i = 0; i < EDGE_IPT; i++) v[i] = p[i];
  }
}

// ---------------------------------------------------------------------------
// fills
// ---------------------------------------------------------------------------
__global__ void k_fill_i32(int* p, int n, int v) {
  int i = blockIdx.x * blockDim.x + threadIdx.x;
  int st = gridDim.x * blockDim.x;
  for (; i < n; i += st) p[i] = v;
}
__global__ void k_fill_f32(float* p, int n, float v) {
  int i = blockIdx.x * blockDim.x + threadIdx.x;
  int st = gridDim.x * blockDim.x;
  for (; i < n; i += st) p[i] = v;
}

// ---------------------------------------------------------------------------
// pos tables: pos1[perm1[i]] = i  (perm entries unique -> race free)
// ---------------------------------------------------------------------------
__global__ void k_scatter_perm(const int* __restrict__ perm1,
                               const int* __restrict__ perm2, int K,
                               int* __restrict__ pos1, int* __restrict__ pos2) {
  int i = blockIdx.x * blockDim.x + threadIdx.x;
  if (i < K) {
    pos1[perm1[i]] = i;
    pos2[perm2[i]] = i;
  }
}

// ---------------------------------------------------------------------------
// node scan pass 1: per-block count of selected nodes (arrays padded, pad = -1)
// ---------------------------------------------------------------------------
__global__ void k_node_pass1(const int* __restrict__ pos1,
                             const int* __restrict__ pos2,
                             int* __restrict__ blockSums) {
  __shared__ int sh[BS];
  int base = (blockIdx.x * BS + threadIdx.x) * NODE_IPT;
  int4 a = *(const int4*)(pos1 + base);
  int4 b = *(const int4*)(pos2 + base);
  int s = ((a.x >= 0) | (b.x >= 0)) + ((a.y >= 0) | (b.y >= 0)) +
          ((a.z >= 0) | (b.z >= 0)) + ((a.w >= 0) | (b.w >= 0));
  int total;
  (void)block_exscan256(s, sh, total);
  if (threadIdx.x == 0) blockSums[blockIdx.x] = total;
}

// ---------------------------------------------------------------------------
// single-block exclusive scan over M block sums, emits grand total
// ---------------------------------------------------------------------------
__global__ void k_scan_sums(int* __restrict__ sums, int M,
                            int* __restrict__ total_out) {
  __shared__ int sh[1024];
  __shared__ int carry_sh;
  int tid = threadIdx.x;
  if (tid == 0) carry_sh = 0;
  __syncthreads();
  for (int basei = 0; basei < M; basei += 1024) {
    int i = basei + tid;
    int v = (i < M) ? sums[i] : 0;
    sh[tid] = v;
    __syncthreads();
    for (int off = 1; off < 1024; off <<= 1) {
      int t = (tid >= off) ? sh[tid - off] : 0;
      __syncthreads();
      sh[tid] += t;
      __syncthreads();
    }
    int incl = sh[tid];
    int chunk_total = sh[1023];
    int carry = carry_sh;
    __syncthreads();
    if (i < M) sums[i] = carry + incl - v;   // exclusive offset
    if (tid == 0) carry_sh = carry + chunk_total;
    __syncthreads();
  }
  if (tid == 0) *total_out = carry_sh;
}

// ---------------------------------------------------------------------------
// node scan pass 3: rank assignment + scatter of perm_pad / new_batch / node_new
// ---------------------------------------------------------------------------
__global__ void k_node_scatter(const int* __restrict__ pos1,
                               const int* __restrict__ pos2,
                               const int* __restrict__ o_batch, int N,
                               const int* __restrict__ blockOffs,
                               int* __restrict__ node_new,
                               int* __restrict__ perm_pad,
                               float* __restrict__ out_batch) {
  __shared__ int sh[BS];
  int base = (blockIdx.x * BS + threadIdx.x) * NODE_IPT;
  int4 a = *(const int4*)(pos1 + base);
  int4 b = *(const int4*)(pos2 + base);
  int sel[NODE_IPT];
  sel[0] = (a.x >= 0) | (b.x >= 0);
  sel[1] = (a.y >= 0) | (b.y >= 0);
  sel[2] = (a.z >= 0) | (b.z >= 0);
  sel[3] = (a.w >= 0) | (b.w >= 0);
  int s = sel[0] + sel[1] + sel[2] + sel[3];
  int total;
  int ex = block_exscan256(s, sh, total);
  int rank = blockOffs[blockIdx.x] + ex;
#pragma unroll
  for (int i = 0; i < NODE_IPT; i++) {
    int n = base + i;
    if (sel[i]) {
      node_new[n] = rank;
      perm_pad[rank] = n;
      if (n < N) out_batch[rank] = (float)o_batch[n];
      rank++;
    } else {
      node_new[n] = -1;
    }
  }
}

// ---------------------------------------------------------------------------
// x-row gather: one wave32 per output row; async global->LDS staging (b128)
// faithful quirk: when in both perms, average x1[pos1] with x2[pos1]
// ---------------------------------------------------------------------------
__global__ void k_xgather(const float* __restrict__ x1,
                          const float* __restrict__ x2,
                          const int* __restrict__ pos1,
                          const int* __restrict__ pos2,
                          const int* __restrict__ perm_pad,
                          float* __restrict__ x3, int R, int D) {
  __shared__ __align__(16) float stage[XG_WAVES][2][128];
  int wave = threadIdx.x >> 5;
  int lane = threadIdx.x & 31;
  int r = blockIdx.x * XG_WAVES + wave;
  if (r >= R) return;
  int p = perm_pad[r];                 // wave-uniform
  float* orow = x3 + (size_t)r * D;
  if (p < 0) {                         // padded row -> zeros
    for (int c = lane * 4; c < D; c += 128) {
      float4 z = {0.f, 0.f, 0.f, 0.f};
      *(float4*)(orow + c) = z;
    }
    return;
  }
  int p1 = pos1[p];                    // wave-uniform
  int p2 = pos2[p];
  for (int c0 = 0; c0 < D; c0 += 128) {
    int c = c0 + lane * 4;
    bool in = (c < D);
    if (p1 >= 0 && p2 >= 0) {
      const float* g1 = x1 + (size_t)p1 * D + c;
      const float* g2 = x2 + (size_t)p1 * D + c;   // quirk: pos1, not pos2
#if defined(__gfx1250__)
      if (in) {
        uint32_t l1 = lds_byte_off(&stage[wave][0][lane * 4]);
        uint32_t l2 = lds_byte_off(&stage[wave][1][lane * 4]);
        uint64_t ga = (uint64_t)(uintptr_t)g1;
        uint64_t gb = (uint64_t)(uintptr_t)g2;
        asm volatile(
            "global_load_async_to_lds_b128 %0, %2, off\n\t"
            "global_load_async_to_lds_b128 %1, %3, off"
            :: "v"(l1), "v"(l2), "v"(ga), "v"(gb)
            : "memory");
      }
      asm volatile("s_wait_asynccnt 0x0" ::: "memory");
      if (in) {
        float4 va = *(const float4*)&stage[wave][0][lane * 4];
        float4 vb = *(const float4*)&stage[wave][1][lane * 4];
        float4 o = {0.5f * (va.x + vb.x), 0.5f * (va.y + vb.y),
                    0.5f * (va.z + vb.z), 0.5f * (va.w + vb.w)};
        *(float4*)(orow + c) = o;
      }
#else
      if (in) {
        float4 va = *(const float4*)g1;
        float4 vb = *(const float4*)g2;
        float4 o = {0.5f * (va.x + vb.x), 0.5f * (va.y + vb.y),
                    0.5f * (va.z + vb.z), 0.5f * (va.w + vb.w)};
        *(float4*)(orow + c) = o;
      }
#endif
    } else {
      // single-source row: direct b128 load->store is optimal (no LDS trip)
      const float* src = (p1 >= 0) ? x1 + (size_t)p1 * D : x2 + (size_t)p2 * D;
      if (in) *(float4*)(orow + c) = *(const float4*)(src + c);
    }
  }
}

// ---------------------------------------------------------------------------
// edge scan pass 1: per-block count of kept edges (vectorized b128 loads)
// ---------------------------------------------------------------------------
__global__ void k_edge_pass1(const int* __restrict__ ei, int E,
                             const int* __restrict__ node_new,
                             int* __restrict__ blockSums) {
  __shared__ int sh[BS];
  int base = (blockIdx.x * BS + threadIdx.x) * EDGE_IPT;
  if (base < E) __builtin_prefetch(ei + E + base, 0, 1);  // dst half ahead
  int s = 0;
  if (base + EDGE_IPT <= E) {
    bool a16 = ((E & 3) == 0);  // base%8==0; dst half needs E%4==0 for b128
    int sv[EDGE_IPT], dv[EDGE_IPT];
    load8(ei + base, true, sv);
    load8(ei + E + base, a16, dv);
#pragma unroll
    for (int i = 0; i < EDGE_IPT; i++)
      s += (node_new[sv[i]] >= 0) & (node_new[dv[i]] >= 0);
  } else {
#pragma unroll
    for (int i = 0; i < EDGE_IPT; i++) {
      int e = base + i;
      if (e < E) s += (node_new[ei[e]] >= 0) & (node_new[ei[E + e]] >= 0);
    }
  }
  int total;
  (void)block_exscan256(s, sh, total);
  if (threadIdx.x == 0) blockSums[blockIdx.x] = total;
}

// ---------------------------------------------------------------------------
// edge scan pass 3: stable compaction of remapped edges + attrs
// ---------------------------------------------------------------------------
__global__ void k_edge_scatter(const int* __restrict__ ei,
                               const float* __restrict__ ea, int E,
                               const int* __restrict__ node_new,
                               const int* __restrict__ blockOffs,
                               float* __restrict__ out_ei0,
                               float* __restrict__ out_ei1,
                               float* __restrict__ out_ea) {
  __shared__ int sh[BS];
  int base = (blockIdx.x * BS + threadIdx.x) * EDGE_IPT;
  if (base < E) __builtin_prefetch(ea + base, 0, 1);
  int av[EDGE_IPT], bv[EDGE_IPT], keep[EDGE_IPT];
  int s = 0;
  if (base + EDGE_IPT <= E) {
    bool a16 = ((E & 3) == 0);
    int sv[EDGE_IPT], dv[EDGE_IPT];
    load8(ei + base, true, sv);
    load8(ei + E + base, a16, dv);
#pragma unroll
    for (int i = 0; i < EDGE_IPT; i++) {
      av[i] = node_new[sv[i]];
      bv[i] = node_new[dv[i]];
      keep[i] = (av[i] >= 0) & (bv[i] >= 0);
      s += keep[i];
    }
  } else {
#pragma unroll
    for (int i = 0; i < EDGE_IPT; i++) {
      int e = base + i;
      keep[i] = 0;
      if (e < E) {
        av[i] = node_new[ei[e]];
        bv[i] = node_new[ei[E + e]];
        keep[i] = (av[i] >= 0) & (bv[i] >= 0);
        s += keep[i];
      }
    }
  }
  int total;
  int ex = block_exscan256(s, sh, total);
  int pos = blockOffs[blockIdx.x] + ex;
#pragma unroll
  for (int i = 0; i < EDGE_IPT; i++) {
    if (keep[i]) {
      out_ei0[pos] = (float)av[i];
      out_ei1[pos] = (float)bv[i];
      out_ea[pos] = ea[base + i];
      pos++;
    }
  }
}

__global__ void k_counts(const int* __restrict__ totals,
                         float* __restrict__ out) {
  if (threadIdx.x == 0) {
    out[0] = (float)totals[0];
    out[1] = (float)totals[1];
  }
}

// ---------------------------------------------------------------------------
// launch
// ---------------------------------------------------------------------------
extern "C" void kernel_launch(void* const* d_in, const int* in_sizes, int n_in,
                              void* d_out, int out_size, void* d_ws,
                              size_t ws_size, hipStream_t stream) {
  const int K = in_sizes[0];
  const int N = in_sizes[5];
  const int D = in_sizes[3] / K;
  const int E = in_sizes[6] / 2;
  const int R = 2 * K;

  const int* perm1 = (const int*)d_in[0];
  const int* perm2 = (const int*)d_in[1];
  const float* x1 = (const float*)d_in[3];
  const float* x2 = (const float*)d_in[4];
  const int* o_batch = (const int*)d_in[5];
  const int* ei = (const int*)d_in[6];
  const float* ea = (const float*)d_in[7];

  float* out = (float*)d_out;
  float* out_x3 = out;                                  // R*D
  float* out_batch = out_x3 + (size_t)R * D;            // R
  float* out_ei = out_batch + R;                        // 2*E
  float* out_ea = out_ei + (size_t)2 * E;               // E
  float* out_counts = out_ea + E;                       // 2

  const int nblocksN = (N + NODE_TILE - 1) / NODE_TILE;
  const int Npad = nblocksN * NODE_TILE;
  const int nblocksE = (E + EDGE_TILE - 1) / EDGE_TILE;

  int* pos1 = (int*)d_ws;
  int* pos2 = pos1 + Npad;
  int* node_new = pos2 + Npad;
  int* perm_pad = node_new + Npad;
  int* nsums = perm_pad + R;
  int* esums = nsums + nblocksN;
  int* totals = esums + nblocksE;

  // init
  k_fill_i32<<<(Npad + BS - 1) / BS, BS, 0, stream>>>(pos1, Npad, -1);
  k_fill_i32<<<(Npad + BS - 1) / BS, BS, 0, stream>>>(pos2, Npad, -1);
  k_fill_i32<<<(R + BS - 1) / BS, BS, 0, stream>>>(perm_pad, R, -1);
  k_fill_f32<<<(R + BS - 1) / BS, BS, 0, stream>>>(out_batch, R, -1.0f);
  k_fill_f32<<<(2 * E + BS - 1) / BS, BS, 0, stream>>>(out_ei, 2 * E, -1.0f);
  k_fill_f32<<<(E + BS - 1) / BS, BS, 0, stream>>>(out_ea, E, 0.0f);

  // membership/position tables
  k_scatter_perm<<<(K + BS - 1) / BS, BS, 0, stream>>>(perm1, perm2, K, pos1,
                                                       pos2);

  // node compaction (deterministic 3-phase scan)
  k_node_pass1<<<nblocksN, BS, 0, stream>>>(pos1, pos2, nsums);
  k_scan_sums<<<1, 1024, 0, stream>>>(nsums, nblocksN, totals + 0);
  k_node_scatter<<<nblocksN, BS, 0, stream>>>(pos1, pos2, o_batch, N, nsums,
                                              node_new, perm_pad, out_batch);

  // x3: one wave per row, async global->LDS staging
  k_xgather<<<(R + XG_WAVES - 1) / XG_WAVES, XG_WAVES * 32, 0, stream>>>(
      x1, x2, pos1, pos2, perm_pad, out_x3, R, D);

  // edge compaction
  k_edge_pass1<<<nblocksE, BS, 0, stream>>>(ei, E, node_new, esums);
  k_scan_sums<<<1, 1024, 0, stream>>>(esums, nblocksE, totals + 1);
  k_edge_scatter<<<nblocksE, BS, 0, stream>>>(ei, ea, E, node_new, esums,
                                              out_ei, out_ei + E, out_ea);

  k_counts<<<1, 32, 0, stream>>>(totals, out_counts);
}